// MGSTGNN_50620484551062
// MI455X (gfx1250) — compile-verified
//
#include <hip/hip_runtime.h>
#include <math.h>

typedef __attribute__((ext_vector_type(2))) float v2f;
typedef __attribute__((ext_vector_type(8))) float v8f;

#define WMMA_F32(A, B, C) \
  __builtin_amdgcn_wmma_f32_16x16x4_f32(false, (A), false, (B), (short)0, (C), false, false)

// Constants: B=32, T=12, N=512, DIN=3, DE=16, HID=64, H=4, hd=16, CIN=67 (pad 68), CS=2, OUT=12
#define NB   32
#define NT   12
#define NN   512
#define NDE  16
#define NHID 64
#define NCP  68   // 67 padded to 68 (multiple of 4 for K-chunks)

// ---------------------------------------------------------------------------
// Precompute W[g][n][slot][i][o] = sum_d node_emb[n][d] * wp[g][d][slot][i][o]
// stored padded: (3, N, 2, 68, 64), rows i>=67 are zero.
__global__ void k_w(const float* __restrict__ ne, const float* __restrict__ wp,
                    float* __restrict__ W) {
  long idx = (long)blockIdx.x * 256 + threadIdx.x;
  const long total = 3L * NN * 2 * NCP * NHID;
  if (idx >= total) return;
  int o = idx % 64; long r = idx / 64;
  int i = r % NCP; r /= NCP;
  int slot = r % 2; r /= 2;
  int n = r % NN; int g = (int)(r / NN);
  float acc = 0.f;
  if (i < 67) {
    for (int d = 0; d < NDE; ++d)
      acc += ne[n * NDE + d] * wp[(((long)(g * NDE + d) * 2 + slot) * 67 + i) * 64 + o];
  }
  W[idx] = acc;
}

// tb[g][t][b][o] = te[b][t][:] @ bp[g][:][o]
__global__ void k_tb(const float* __restrict__ te, const float* __restrict__ bp,
                     float* __restrict__ TB) {
  int idx = blockIdx.x * 256 + threadIdx.x;
  if (idx >= 3 * NT * NB * NHID) return;
  int o = idx % 64; int r = idx / 64;
  int b = r % NB; r /= NB;
  int t = r % NT; int g = r / NT;
  float acc = 0.f;
  for (int d = 0; d < NDE; ++d)
    acc += te[(b * NT + t) * NDE + d] * bp[(g * NDE + d) * 64 + o];
  TB[idx] = acc;
}

// emb[g][b][n][:] = LN_{16}(node_emb[n] + te[b][t]) * glg[g] + glb[g], eps 1e-12
__global__ void k_emb(const float* __restrict__ ne, const float* __restrict__ te,
                      const float* __restrict__ glg, const float* __restrict__ glb,
                      float* __restrict__ E, int t) {
  int idx = blockIdx.x * 256 + threadIdx.x;   // rows: 3*B*N
  if (idx >= 3 * NB * NN) return;
  int n = idx % NN; int r = idx / NN;
  int b = r % NB; int g = r / NB;
  float v[NDE]; float mu = 0.f;
  for (int d = 0; d < NDE; ++d) { v[d] = ne[n * NDE + d] + te[(b * NT + t) * NDE + d]; mu += v[d]; }
  mu *= (1.f / NDE);
  float var = 0.f;
  for (int d = 0; d < NDE; ++d) { float q = v[d] - mu; var += q * q; }
  var *= (1.f / NDE);
  float inv = rsqrtf(var + 1e-12f);
  for (int d = 0; d < NDE; ++d)
    E[(long)idx * NDE + d] = (v[d] - mu) * inv * glg[g * NDE + d] + glb[g * NDE + d];
}

__global__ void k_zero(float* p, long nElem) {
  long i = (long)blockIdx.x * 256 + threadIdx.x;
  if (i < nElem) p[i] = 0.f;
}

// ias[b][n][0:3]=x[b][t][n], [3:67]=h_prev[b][n], [67]=0
__global__ void k_build_ias(const float* __restrict__ x, const float* __restrict__ h,
                            float* __restrict__ ias, int t) {
  int idx = blockIdx.x * 256 + threadIdx.x;
  if (idx >= NB * NN * NCP) return;
  int i = idx % NCP; int r = idx / NCP; int n = r % NN; int b = r / NN;
  float v;
  if (i < 3)       v = x[((long)(b * NT + t) * NN + n) * 3 + i];
  else if (i < 67) v = h[((long)b * NN + n) * 64 + (i - 3)];
  else             v = 0.f;
  ias[idx] = v;
}

// cand: state scaled by z = sigmoid(zpre)
__global__ void k_build_cand(const float* __restrict__ x, const float* __restrict__ h,
                             const float* __restrict__ zp, float* __restrict__ cand, int t) {
  int idx = blockIdx.x * 256 + threadIdx.x;
  if (idx >= NB * NN * NCP) return;
  int i = idx % NCP; int r = idx / NCP; int n = r % NN; int b = r / NN;
  float v;
  if (i < 3)       v = x[((long)(b * NT + t) * NN + n) * 3 + i];
  else if (i < 67) {
    long e = ((long)b * NN + n) * 64 + (i - 3);
    float z = 1.f / (1.f + __expf(-zp[e]));
    v = z * h[e];
  } else v = 0.f;
  cand[idx] = v;
}

// ---------------------------------------------------------------------------
// Fused graph attention: Y[b] = softmax(E[g,b] @ E[g,b]^T) @ X[b]
// fp32 WMMA 16x16x4; no per-chunk reductions: exp applied directly (scores of
// LN'd 16-dim embeddings are small), denominator gathered by a ones-column in
// the V tile so the P@V WMMA accumulates it for free. Block = 4 waves, each
// owns a 16-row tile; grid = (N/64, B).
__global__ void __launch_bounds__(128) k_graph_attn(
    const float* __restrict__ E,   // (3,B,N,16)
    const float* __restrict__ X,   // (B,N,68)
    float* __restrict__ Y,         // (B,N,68)
    int g) {
  __shared__ float lds_e[16][16];
  __shared__ float lds_v[16][80];      // cols 0..67 data, col 68 = 1.0, 69..79 = 0
  __shared__ float lds_p[4][16][17];   // per-wave P staging (C->A layout hop)

  const int b = blockIdx.y;
  const int row0 = blockIdx.x * 64;
  const int tid = threadIdx.x;
  const int wave = tid >> 5;
  const int lane = tid & 31;
  const int l16 = lane & 15;
  const int half = lane >> 4;
  const int myrow = row0 + wave * 16 + l16;

  const float* Eb = E + ((long)g * NB + b) * NN * NDE;
  const float* Xb = X + (long)b * NN * NCP;

  // A fragments: this wave's 16 emb rows (K=16 split into 4 chunks of 4)
  v2f aA[4];
#pragma unroll
  for (int kk = 0; kk < 4; ++kk) {
    aA[kk].x = Eb[myrow * NDE + kk * 4 + 2 * half];
    aA[kk].y = Eb[myrow * NDE + kk * 4 + 2 * half + 1];
  }

  // init pad columns once: col 68 = ones (rowsum trick), 69..79 = 0.
  // chunk copies below never touch cols >= 68, so this persists.
  for (int i = tid; i < 16 * 12; i += 128) {
    int rr = i / 12, cc = 68 + i % 12;
    lds_v[rr][cc] = (cc == 68) ? 1.f : 0.f;
  }

  v8f acc[5] = {};

  for (int mc = 0; mc < NN / 16; ++mc) {
    __syncthreads();   // protect lds_e/lds_v against previous iteration's reads
    {
      const float2* se = (const float2*)(Eb + (long)mc * 16 * NDE);   // 256 floats contiguous
      ((float2*)lds_e)[tid] = se[tid];
    }
    {
      const float2* sv = (const float2*)(Xb + (long)mc * 16 * NCP);   // 1088 floats contiguous
      for (int i = tid; i < 544; i += 128) {
        int rr = i / 34, cc = (i % 34) * 2;
        *(float2*)&lds_v[rr][cc] = sv[i];
      }
    }
    __syncthreads();

    // Score tile S = A(16x16) @ E_chunk^T(16x16)
    v8f c = {};
#pragma unroll
    for (int kk = 0; kk < 4; ++kk) {
      v2f bf;
      bf.x = lds_e[l16][kk * 4 + 2 * half];
      bf.y = lds_e[l16][kk * 4 + 2 * half + 1];
      c = WMMA_F32(aA[kk], bf, c);
    }

    // P = exp(S), staged through wave-private LDS (same-wave DS ops are
    // in-order; no block barrier needed)
#pragma unroll
    for (int r = 0; r < 8; ++r)
      lds_p[wave][r + 8 * half][l16] = __expf(c[r]);

    // O += P @ V_chunk  (5 col tiles; tile 4 includes the ones-column)
#pragma unroll
    for (int kk = 0; kk < 4; ++kk) {
      v2f ap;
      ap.x = lds_p[wave][l16][kk * 4 + 2 * half];
      ap.y = lds_p[wave][l16][kk * 4 + 2 * half + 1];
#pragma unroll
      for (int ct = 0; ct < 5; ++ct) {
        v2f bv;
        bv.x = lds_v[kk * 4 + 2 * half][ct * 16 + l16];
        bv.y = lds_v[kk * 4 + 2 * half + 1][ct * 16 + l16];
        acc[ct] = WMMA_F32(ap, bv, acc[ct]);
      }
    }
  }

  // Row denominators live in acc[4] at column 68 (l16 == 4 of each half)
#pragma unroll
  for (int r = 0; r < 8; ++r) {
    float rs = __shfl(acc[4][r], half * 16 + 4);
    float inv = 1.f / rs;
    int row = row0 + wave * 16 + r + 8 * half;
#pragma unroll
    for (int ct = 0; ct < 5; ++ct) {
      int col = ct * 16 + l16;
      if (col < NCP)
        Y[((long)b * NN + row) * NCP + col] = acc[ct][r] * inv;
    }
  }
}

// ---------------------------------------------------------------------------
// Per-node GEMM: G[b][n][o] = X[b][n][:]@W[g][n][0] + XG[b][n][:]@W[g][n][1] + TB[g][t][b][o]
// One node per block; wave = b-tile (2x16 rows); 4 column tiles; K=68.
__global__ void __launch_bounds__(64) k_gout(
    const float* __restrict__ X, const float* __restrict__ XG,
    const float* __restrict__ W, const float* __restrict__ TB,
    float* __restrict__ G, int g, int t) {
  const int n = blockIdx.x;
  const int wave = threadIdx.x >> 5;
  const int lane = threadIdx.x & 31;
  const int l16 = lane & 15, half = lane >> 4;
  const float* Wg = W + ((long)g * NN + n) * 2 * NCP * NHID;

  v8f c[4] = {};
  for (int slot = 0; slot < 2; ++slot) {
    const float* Arow = (slot ? XG : X) + ((long)(wave * 16 + l16) * NN + n) * NCP;
    const float* Wm = Wg + (long)slot * NCP * NHID;
    for (int kk = 0; kk < NCP / 4; ++kk) {
      v2f a;
      a.x = Arow[kk * 4 + 2 * half];
      a.y = Arow[kk * 4 + 2 * half + 1];
#pragma unroll
      for (int ct = 0; ct < 4; ++ct) {
        v2f bw;
        bw.x = Wm[(kk * 4 + 2 * half) * 64 + ct * 16 + l16];
        bw.y = Wm[(kk * 4 + 2 * half + 1) * 64 + ct * 16 + l16];
        c[ct] = WMMA_F32(a, bw, c[ct]);
      }
    }
  }
#pragma unroll
  for (int ct = 0; ct < 4; ++ct)
#pragma unroll
    for (int r = 0; r < 8; ++r) {
      int bb = wave * 16 + r + 8 * half;
      int o = ct * 16 + l16;
      G[((long)bb * NN + n) * 64 + o] = c[ct][r] + TB[(((long)g * NT + t) * NB + bb) * 64 + o];
    }
}

// ---------------------------------------------------------------------------
// q = LN64(G); 4-head attention over K stored states; G += o. One wave per row.
__global__ void __launch_bounds__(256) k_state_attn(
    float* __restrict__ G, const float* __restrict__ Hs,
    const float* __restrict__ olg, const float* __restrict__ olb, int g, int K) {
  const int wave = threadIdx.x >> 5, lane = threadIdx.x & 31;
  const long row = (long)blockIdx.x * 8 + wave;   // B*N = 16384 rows
  float g0 = G[row * 64 + lane];
  float g1 = G[row * 64 + lane + 32];
  float s = g0 + g1;
  s += __shfl_xor(s, 1); s += __shfl_xor(s, 2); s += __shfl_xor(s, 4);
  s += __shfl_xor(s, 8); s += __shfl_xor(s, 16);
  float mu = s * (1.f / 64.f);
  float d0 = g0 - mu, d1 = g1 - mu;
  float v = d0 * d0 + d1 * d1;
  v += __shfl_xor(v, 1); v += __shfl_xor(v, 2); v += __shfl_xor(v, 4);
  v += __shfl_xor(v, 8); v += __shfl_xor(v, 16);
  float inv = rsqrtf(v * (1.f / 64.f) + 1e-5f);
  float q0 = d0 * inv * olg[g * 64 + lane] + olb[g * 64 + lane];
  float q1 = d1 * inv * olg[g * 64 + lane + 32] + olb[g * 64 + lane + 32];

  float st0[13], st1[13], sc0[13], sc1[13];
#pragma unroll
  for (int k = 0; k < 13; ++k) {
    if (k < K) {
      st0[k] = Hs[((long)k * NB * NN + row) * 64 + lane];
      st1[k] = Hs[((long)k * NB * NN + row) * 64 + lane + 32];
      float p0 = q0 * st0[k], p1 = q1 * st1[k];
      p0 += __shfl_xor(p0, 1); p0 += __shfl_xor(p0, 2);
      p0 += __shfl_xor(p0, 4); p0 += __shfl_xor(p0, 8);
      p1 += __shfl_xor(p1, 1); p1 += __shfl_xor(p1, 2);
      p1 += __shfl_xor(p1, 4); p1 += __shfl_xor(p1, 8);
      sc0[k] = p0 * 0.25f; sc1[k] = p1 * 0.25f;   // / sqrt(hd=16)
    } else { st0[k] = 0.f; st1[k] = 0.f; sc0[k] = -1e30f; sc1[k] = -1e30f; }
  }
  float m0 = -1e30f, m1 = -1e30f;
#pragma unroll
  for (int k = 0; k < 13; ++k) { m0 = fmaxf(m0, sc0[k]); m1 = fmaxf(m1, sc1[k]); }
  float s0 = 0.f, s1 = 0.f;
#pragma unroll
  for (int k = 0; k < 13; ++k) {
    sc0[k] = __expf(sc0[k] - m0); s0 += sc0[k];
    sc1[k] = __expf(sc1[k] - m1); s1 += sc1[k];
  }
  float o0 = 0.f, o1 = 0.f;
#pragma unroll
  for (int k = 0; k < 13; ++k) { o0 += sc0[k] * st0[k]; o1 += sc1[k] * st1[k]; }
  G[row * 64 + lane]      = g0 + o0 / s0;
  G[row * 64 + lane + 32] = g1 + o1 / s1;
}

// h_new = sigmoid(r) * h_prev + (1 - sigmoid(r)) * tanh(hc)
__global__ void k_combine(const float* __restrict__ Gr, const float* __restrict__ Ghc,
                          const float* __restrict__ Hprev, float* __restrict__ Hnew) {
  long i = (long)blockIdx.x * 256 + threadIdx.x;
  if (i >= (long)NB * NN * 64) return;
  float r = 1.f / (1.f + __expf(-Gr[i]));
  float hc = tanhf(Ghc[i]);
  Hnew[i] = r * Hprev[i] + (1.f - r) * hc;
}

// pred[b][o][n] = sum_{c,h} Hs[T-1+c][b][n][h] * pw[o][c][h] + pb[o]
__global__ void k_pred(const float* __restrict__ Hs, const float* __restrict__ pw,
                       const float* __restrict__ pb, float* __restrict__ out) {
  int idx = blockIdx.x * 256 + threadIdx.x;   // (b*12 + o)*512 + n
  if (idx >= NB * 12 * NN) return;
  int n = idx % NN; int r = idx / NN; int o = r % 12; int b = r / 12;
  float acc = pb[o];
  for (int cc = 0; cc < 2; ++cc) {
    const float* h = Hs + ((long)(NT - 1 + cc) * NB * NN + (long)b * NN + n) * 64;
    for (int hh = 0; hh < 64; ++hh)
      acc += h[hh] * pw[(o * 2 + cc) * 64 + hh];
  }
  out[idx] = acc;
}

// ---------------------------------------------------------------------------
extern "C" void kernel_launch(void* const* d_in, const int* in_sizes, int n_in,
                              void* d_out, int out_size, void* d_ws, size_t ws_size,
                              hipStream_t stream) {
  const float* x   = (const float*)d_in[0];
  const float* ne  = (const float*)d_in[1];
  const float* te  = (const float*)d_in[2];
  const float* wp  = (const float*)d_in[3];
  const float* bp  = (const float*)d_in[4];
  const float* glg = (const float*)d_in[5];
  const float* glb = (const float*)d_in[6];
  const float* olg = (const float*)d_in[7];
  const float* olb = (const float*)d_in[8];
  const float* pw  = (const float*)d_in[9];
  const float* pb  = (const float*)d_in[10];
  float* out = (float*)d_out;

  float* ws = (float*)d_ws;
  long off = 0;
  float* W    = ws + off; off += 3L * NN * 2 * NCP * NHID;   // 13.37 M
  float* TB   = ws + off; off += 3L * NT * NB * NHID;
  float* E    = ws + off; off += 3L * NB * NN * NDE;
  float* Hs   = ws + off; off += (long)(NT + 1) * NB * NN * NHID;
  float* IAS  = ws + off; off += (long)NB * NN * NCP;
  float* CAND = ws + off; off += (long)NB * NN * NCP;
  float* XG2  = ws + off; off += (long)NB * NN * NCP;
  float* G0   = ws + off; off += (long)NB * NN * NHID;
  float* G1   = ws + off; off += (long)NB * NN * NHID;
  float* G2   = ws + off; off += (long)NB * NN * NHID;

  const long wTotal = 3L * NN * 2 * NCP * NHID;
  k_w<<<dim3((unsigned)((wTotal + 255) / 256)), 256, 0, stream>>>(ne, wp, W);
  k_tb<<<(3 * NT * NB * NHID + 255) / 256, 256, 0, stream>>>(te, bp, TB);
  k_zero<<<(unsigned)(((long)NB * NN * NHID + 255) / 256), 256, 0, stream>>>(Hs, (long)NB * NN * NHID);

  for (int t = 0; t < NT; ++t) {
    float* Hprev = Hs + (long)t * NB * NN * NHID;
    float* Hnew  = Hs + (long)(t + 1) * NB * NN * NHID;
    k_emb<<<(3 * NB * NN + 255) / 256, 256, 0, stream>>>(ne, te, glg, glb, E, t);
    k_build_ias<<<(NB * NN * NCP + 255) / 256, 256, 0, stream>>>(x, Hprev, IAS, t);
    for (int g = 0; g < 2; ++g) {
      float* Gg = g ? G1 : G0;
      k_graph_attn<<<dim3(NN / 64, NB), 128, 0, stream>>>(E, IAS, XG2, g);
      k_gout<<<NN, 64, 0, stream>>>(IAS, XG2, W, TB, Gg, g, t);
      k_state_attn<<<NB * NN / 8, 256, 0, stream>>>(Gg, Hs, olg, olb, g, t + 1);
    }
    k_build_cand<<<(NB * NN * NCP + 255) / 256, 256, 0, stream>>>(x, Hprev, G0, CAND, t);
    k_graph_attn<<<dim3(NN / 64, NB), 128, 0, stream>>>(E, CAND, XG2, 2);
    k_gout<<<NN, 64, 0, stream>>>(CAND, XG2, W, TB, G2, 2, t);
    k_state_attn<<<NB * NN / 8, 256, 0, stream>>>(G2, Hs, olg, olb, 2, t + 1);
    k_combine<<<(unsigned)(((long)NB * NN * NHID + 255) / 256), 256, 0, stream>>>(G1, G2, Hprev, Hnew);
  }
  k_pred<<<(NB * 12 * NN + 255) / 256, 256, 0, stream>>>(Hs, pw, pb, out);
}